// GraphVAEWithHGT_50723563766183
// MI455X (gfx1250) — compile-verified
//
#include <hip/hip_runtime.h>
#include <hip/hip_bf16.h>

// ---------------------------------------------------------------------------
// Problem constants (match reference)
// ---------------------------------------------------------------------------
#define kN    48000
#define kNT   16000
#define kE    512000
#define kDIN  256
#define kD    64
#define kH    4
#define kDK   16
#define kR    6
#define kT    3
#define kL    2

#define OFF_MU  ((size_t)12288000)   // 3*NT*DIN
#define OFF_LV  ((size_t)15360000)   // OFF_MU + N*D

// workspace float offsets
#define WS_H      ((size_t)0)
#define WS_Q      ((size_t)3072000)
#define WS_K      ((size_t)6144000)
#define WS_V      ((size_t)9216000)
#define WS_SCORE  ((size_t)12288000)  // E*H = 2,048,000
#define WS_MKEY   ((size_t)14336000)  // N*H  (uint keys)
#define WS_Z      ((size_t)14528000)  // N*H
#define WS_OUTUN  ((size_t)14720000)  // N*D
#define WS_ZB     WS_Q                // reuse Q region for latent z after layers
// total ws use: 17,792,000 floats (~71.2 MB)

typedef __attribute__((ext_vector_type(2))) float v2f;
typedef __attribute__((ext_vector_type(8))) float v8f;

__device__ __forceinline__ v8f wmma4(v2f a, v2f b, v8f c) {
  // D = A(16x4 f32) * B(4x16 f32) + C(16x16 f32)
  return __builtin_amdgcn_wmma_f32_16x16x4_f32(false, a, false, b, (short)0, c,
                                               false, false);
}

__device__ __forceinline__ unsigned fkey(float f) {
  unsigned u = __float_as_uint(f);
  return (u & 0x80000000u) ? ~u : (u | 0x80000000u);
}
__device__ __forceinline__ float funkey(unsigned k) {
  unsigned u = (k & 0x80000000u) ? (k & 0x7FFFFFFFu) : ~k;
  return __uint_as_float(u);
}

// Stage a [rows x cols] row-major weight matrix into LDS in K-pair-interleaved
// v2f layout: wl2[p*cols + n] = { W[2p][n], W[2p+1][n] }.
// B-fragment for K-step k, half 'hi', column c is then a single b64 load:
//   wl2[(k/2 + hi)*cols + c]
__device__ __forceinline__ void stage_w_pairs(v2f* wl2, const float* W,
                                              int pairs, int cols) {
  for (int i = threadIdx.x; i < pairs * cols; i += 256) {
    int p = i / cols, n = i - p * cols;
    v2f t = {W[(2 * p) * cols + n], W[(2 * p + 1) * cols + n]};
    wl2[i] = t;
  }
}

// ---------------------------------------------------------------------------
// Encoder: per-type  X[16000,256] @ encW[t][256,64] + encb[t]  -> h
// grid = 3*125, block = 256 (8 waves, each wave does a 16x64 output tile)
// ---------------------------------------------------------------------------
__global__ __launch_bounds__(256) void enc_kernel(
    const float* __restrict__ x0, const float* __restrict__ x1,
    const float* __restrict__ x2, const float* __restrict__ encW,
    const float* __restrict__ encb, float* __restrict__ h) {
  __shared__ v2f wl2[(kDIN / 2) * kD];  // 64 KB, pair-interleaved
  int blk = blockIdx.x;
  int t = blk / 125;
  int rowBlk = blk % 125;
  const float* X = (t == 0) ? x0 : ((t == 1) ? x1 : x2);
  const float* W = encW + (size_t)t * kDIN * kD;
  stage_w_pairs(wl2, W, kDIN / 2, kD);
  __syncthreads();

  int wave = threadIdx.x >> 5, lane = threadIdx.x & 31;
  int hi = lane >> 4, ln = lane & 15;
  int arow = rowBlk * 128 + wave * 16 + ln;  // A-matrix row for this lane
  v8f c0 = {0,0,0,0,0,0,0,0}, c1 = c0, c2 = c0, c3 = c0;

  for (int k = 0; k < kDIN; k += 4) {
    v2f a = *(const v2f*)(X + (size_t)arow * kDIN + k + 2 * hi);
    const v2f* wp = wl2 + (k / 2 + hi) * kD;
    v2f b0 = wp[ln];
    v2f b1 = wp[16 + ln];
    v2f b2 = wp[32 + ln];
    v2f b3 = wp[48 + ln];
    c0 = wmma4(a, b0, c0); c1 = wmma4(a, b1, c1);
    c2 = wmma4(a, b2, c2); c3 = wmma4(a, b3, c3);
  }
  const float* bias = encb + t * kD;
  int mrow = t * kNT + rowBlk * 128 + wave * 16 + 8 * hi;
#pragma unroll
  for (int r = 0; r < 8; ++r) {
    size_t o = (size_t)(mrow + r) * kD;
    h[o + ln]      = c0[r] + bias[ln];
    h[o + 16 + ln] = c1[r] + bias[16 + ln];
    h[o + 32 + ln] = c2[r] + bias[32 + ln];
    h[o + 48 + ln] = c3[r] + bias[48 + ln];
  }
}

// ---------------------------------------------------------------------------
// QKV: per-type  h[16000,64] @ {q,k,v}W[l][t][64,64] + bias
// grid = 3*125, block = 256; h block (32 KB) + current weight (16 KB) in LDS
// ---------------------------------------------------------------------------
__global__ __launch_bounds__(256) void qkv_kernel(
    const float* __restrict__ h, const float* __restrict__ qW,
    const float* __restrict__ qb, const float* __restrict__ kW,
    const float* __restrict__ kb, const float* __restrict__ vW,
    const float* __restrict__ vb, int l, float* __restrict__ Qo,
    float* __restrict__ Ko, float* __restrict__ Vo) {
  __shared__ float hl[128 * kD];        // 32 KB
  __shared__ v2f wl2[(kD / 2) * kD];    // 16 KB, pair-interleaved
  int blk = blockIdx.x;
  int t = blk / 125;
  int rowBlk = blk % 125;
  int rowBase = t * kNT + rowBlk * 128;
  for (int i = threadIdx.x; i < 128 * kD; i += 256)
    hl[i] = h[(size_t)rowBase * kD + i];

  int wave = threadIdx.x >> 5, lane = threadIdx.x & 31;
  int hi = lane >> 4, ln = lane & 15;
  const float* lh = hl + (wave * 16 + ln) * kD;

#pragma unroll
  for (int w = 0; w < 3; ++w) {
    const float* W = ((w == 0) ? qW : (w == 1) ? kW : vW) +
                     (size_t)(l * kT + t) * kD * kD;
    const float* B = ((w == 0) ? qb : (w == 1) ? kb : vb) +
                     (size_t)(l * kT + t) * kD;
    float* O = (w == 0) ? Qo : (w == 1) ? Ko : Vo;
    __syncthreads();  // hl staged (w==0) / previous pass done reading wl2
    stage_w_pairs(wl2, W, kD / 2, kD);
    __syncthreads();

    v8f c0 = {0,0,0,0,0,0,0,0}, c1 = c0, c2 = c0, c3 = c0;
    for (int k = 0; k < kD; k += 4) {
      v2f a = *(const v2f*)(lh + k + 2 * hi);
      const v2f* wp = wl2 + (k / 2 + hi) * kD;
      v2f b0 = wp[ln];
      v2f b1 = wp[16 + ln];
      v2f b2 = wp[32 + ln];
      v2f b3 = wp[48 + ln];
      c0 = wmma4(a, b0, c0); c1 = wmma4(a, b1, c1);
      c2 = wmma4(a, b2, c2); c3 = wmma4(a, b3, c3);
    }
    int mrow = rowBase + wave * 16 + 8 * hi;
#pragma unroll
    for (int r = 0; r < 8; ++r) {
      size_t o = (size_t)(mrow + r) * kD;
      O[o + ln]      = c0[r] + B[ln];
      O[o + 16 + ln] = c1[r] + B[16 + ln];
      O[o + 32 + ln] = c2[r] + B[32 + ln];
      O[o + 48 + ln] = c3[r] + B[48 + ln];
    }
  }
}

// ---------------------------------------------------------------------------
// zero-fill scratch region (mkey keys: 0 == -inf under the monotone mapping)
// ---------------------------------------------------------------------------
__global__ __launch_bounds__(256) void fill0_kernel(float* __restrict__ p, int n) {
  int i = blockIdx.x * blockDim.x + threadIdx.x;
  if (i < n) p[i] = 0.0f;
}

// ---------------------------------------------------------------------------
// Edge pass A: scores[e,h] + segment max over dst (monotone-uint atomicMax)
// one thread per (edge, head); E*H = 2,048,000 threads
// ---------------------------------------------------------------------------
__global__ __launch_bounds__(256) void edgeA_kernel(
    const int* __restrict__ ei, const int* __restrict__ et,
    const int* __restrict__ esign, const float* __restrict__ dist,
    const float* __restrict__ Q, const float* __restrict__ Kb,
    const float* __restrict__ rq, const float* __restrict__ rk,
    const float* __restrict__ skn, const float* __restrict__ rb,
    const float* __restrict__ da_p, const float* __restrict__ dt_p,
    float* __restrict__ score, unsigned* __restrict__ mkey) {
  int gid = blockIdx.x * blockDim.x + threadIdx.x;
  if (gid >= kE * kH) return;
  int e = gid >> 2, hh = gid & 3;
  int s = ei[e], d = ei[kE + e];
  int rel = et[e], sg = esign[e];
  int sidx = (sg == -1) ? 0 : ((sg == 1) ? 1 : 2);

  const float4* q4  = (const float4*)(Q  + (size_t)d * kD + hh * kDK);
  const float4* k4  = (const float4*)(Kb + (size_t)s * kD + hh * kDK);
  const float4* rq4 = (const float4*)(rq + (rel * kH + hh) * kDK);
  const float4* rk4 = (const float4*)(rk + (rel * kH + hh) * kDK);
  const float4* sk4 = (const float4*)(skn + hh * kDK);

  float acc = 0.0f;
#pragma unroll
  for (int i = 0; i < 4; ++i) {
    float4 qv = q4[i], kv = k4[i], rqv = rq4[i], rkv = rk4[i];
    float sx, sy, sz, sw;
    if (sidx == 2) {
      float4 sv = sk4[i];
      sx = sv.x; sy = sv.y; sz = sv.z; sw = sv.w;
    } else {
      float c = (sidx == 0) ? -1.0f : 1.0f;
      sx = c; sy = c; sz = c; sw = c;
    }
    acc += qv.x * rqv.x * kv.x * rkv.x * sx;
    acc += qv.y * rqv.y * kv.y * rkv.y * sy;
    acc += qv.z * rqv.z * kv.z * rkv.z * sz;
    acc += qv.w * rqv.w * kv.w * rkv.w * sw;
  }
  float da = da_p[0], dt = dt_p[0];
  float phi = da * expf(-dist[e] / (dt + 1e-9f));
  float sc = acc * 0.25f + rb[rel * kH + hh] + phi;  // 1/sqrt(16) = 0.25
  score[gid] = sc;
  atomicMax(mkey + (size_t)d * kH + hh, fkey(sc));
}

// ---------------------------------------------------------------------------
// Edge pass B: es = exp(score - max); Z += es; out_un += es * v_eff
// ---------------------------------------------------------------------------
__global__ __launch_bounds__(256) void edgeB_kernel(
    const int* __restrict__ ei, const int* __restrict__ et,
    const int* __restrict__ esign, const float* __restrict__ Vb,
    const float* __restrict__ rv, const float* __restrict__ svn,
    const float* __restrict__ score, const unsigned* __restrict__ mkey,
    float* __restrict__ Z, float* __restrict__ outun) {
  int gid = blockIdx.x * blockDim.x + threadIdx.x;
  if (gid >= kE * kH) return;
  int e = gid >> 2, hh = gid & 3;
  int s = ei[e], d = ei[kE + e];
  int rel = et[e], sg = esign[e];
  int sidx = (sg == -1) ? 0 : ((sg == 1) ? 1 : 2);

  float m = funkey(mkey[(size_t)d * kH + hh]);
  float es = expf(score[gid] - m);
  atomicAdd(Z + (size_t)d * kH + hh, es);

  const float4* v4  = (const float4*)(Vb + (size_t)s * kD + hh * kDK);
  const float4* rv4 = (const float4*)(rv + (rel * kH + hh) * kDK);
  const float4* sv4 = (const float4*)(svn + hh * kDK);
  float* ob = outun + (size_t)d * kD + hh * kDK;
#pragma unroll
  for (int i = 0; i < 4; ++i) {
    float4 vv = v4[i], rvv = rv4[i];
    float sx, sy, sz, sw;
    if (sidx == 2) {
      float4 sv = sv4[i];
      sx = sv.x; sy = sv.y; sz = sv.z; sw = sv.w;
    } else {
      float c = (sidx == 0) ? -1.0f : 1.0f;
      sx = c; sy = c; sz = c; sw = c;
    }
    atomicAdd(ob + i * 4 + 0, es * vv.x * rvv.x * sx);
    atomicAdd(ob + i * 4 + 1, es * vv.y * rvv.y * sy);
    atomicAdd(ob + i * 4 + 2, es * vv.z * rvv.z * sz);
    atomicAdd(ob + i * 4 + 3, es * vv.w * rvv.w * sw);
  }
}

// ---------------------------------------------------------------------------
// Node finalize: out = out_un/(Z+1e-9); skip-mix; LayerNorm. Wave per node.
// ---------------------------------------------------------------------------
__global__ __launch_bounds__(256) void nodeFin_kernel(
    const float* __restrict__ outun, const float* __restrict__ Z,
    float* __restrict__ h, const float* __restrict__ skip_l,
    const float* __restrict__ lg, const float* __restrict__ lb) {
  int wid = (blockIdx.x * blockDim.x + threadIdx.x) >> 5;
  int lane = threadIdx.x & 31;
  if (wid >= kN) return;
  int t = wid / kNT;
  int d0 = lane * 2;
  int hh = d0 >> 4;

  float2 o = *(const float2*)(outun + (size_t)wid * kD + d0);
  float z = Z[(size_t)wid * kH + hh] + 1e-9f;
  o.x /= z; o.y /= z;
  float2 hv = *(const float2*)(h + (size_t)wid * kD + d0);
  float alpha = 1.0f / (1.0f + expf(-skip_l[t]));
  float px = alpha * o.x + (1.0f - alpha) * hv.x;
  float py = alpha * o.y + (1.0f - alpha) * hv.y;

  float sum = px + py;
  float sq = px * px + py * py;
#pragma unroll
  for (int m = 16; m >= 1; m >>= 1) {
    sum += __shfl_xor(sum, m, 32);
    sq += __shfl_xor(sq, m, 32);
  }
  float mean = sum * (1.0f / kD);
  float var = sq * (1.0f / kD) - mean * mean;
  float rs = rsqrtf(var + 1e-5f);
  float2 g = *(const float2*)(lg + t * kD + d0);
  float2 b = *(const float2*)(lb + t * kD + d0);
  float2 hn;
  hn.x = (px - mean) * rs * g.x + b.x;
  hn.y = (py - mean) * rs * g.y + b.y;
  *(float2*)(h + (size_t)wid * kD + d0) = hn;
}

// ---------------------------------------------------------------------------
// mu/logvar/z: h[48000,64] @ muW/lvW [64,64]; z = mu + eps*exp(0.5*lv)
// grid = 375, block = 256; h (32 KB) + both weights interleaved (32 KB) in LDS
// ---------------------------------------------------------------------------
__global__ __launch_bounds__(256) void mulvz_kernel(
    const float* __restrict__ h, const float* __restrict__ muW,
    const float* __restrict__ mub, const float* __restrict__ lvW,
    const float* __restrict__ lvb, const float* __restrict__ eps,
    float* __restrict__ out, float* __restrict__ zb) {
  __shared__ float hl[128 * kD];         // 32 KB
  __shared__ v2f wmu[(kD / 2) * kD];     // 16 KB
  __shared__ v2f wlv[(kD / 2) * kD];     // 16 KB
  int rowBase = blockIdx.x * 128;
  for (int i = threadIdx.x; i < 128 * kD; i += 256)
    hl[i] = h[(size_t)rowBase * kD + i];
  stage_w_pairs(wmu, muW, kD / 2, kD);
  stage_w_pairs(wlv, lvW, kD / 2, kD);
  __syncthreads();

  int wave = threadIdx.x >> 5, lane = threadIdx.x & 31;
  int hi = lane >> 4, ln = lane & 15;
  const float* lh = hl + (wave * 16 + ln) * kD;

  v8f m0 = {0,0,0,0,0,0,0,0}, m1 = m0, m2 = m0, m3 = m0;
  v8f l0 = m0, l1 = m0, l2 = m0, l3 = m0;
  for (int k = 0; k < kD; k += 4) {
    v2f a = *(const v2f*)(lh + k + 2 * hi);
    const v2f* mp = wmu + (k / 2 + hi) * kD;
    const v2f* lp = wlv + (k / 2 + hi) * kD;
    v2f bm0 = mp[ln], bm1 = mp[16 + ln], bm2 = mp[32 + ln], bm3 = mp[48 + ln];
    v2f bl0 = lp[ln], bl1 = lp[16 + ln], bl2 = lp[32 + ln], bl3 = lp[48 + ln];
    m0 = wmma4(a, bm0, m0); m1 = wmma4(a, bm1, m1);
    m2 = wmma4(a, bm2, m2); m3 = wmma4(a, bm3, m3);
    l0 = wmma4(a, bl0, l0); l1 = wmma4(a, bl1, l1);
    l2 = wmma4(a, bl2, l2); l3 = wmma4(a, bl3, l3);
  }
  int mrow = rowBase + wave * 16 + 8 * hi;
#pragma unroll
  for (int r = 0; r < 8; ++r) {
    size_t base = (size_t)(mrow + r) * kD;
#pragma unroll
    for (int j = 0; j < 4; ++j) {
      float mu_v = ((j == 0) ? m0[r] : (j == 1) ? m1[r] : (j == 2) ? m2[r] : m3[r]) +
                   mub[j * 16 + ln];
      float lv_v = ((j == 0) ? l0[r] : (j == 1) ? l1[r] : (j == 2) ? l2[r] : l3[r]) +
                   lvb[j * 16 + ln];
      size_t o = base + j * 16 + ln;
      out[OFF_MU + o] = mu_v;
      out[OFF_LV + o] = lv_v;
      zb[o] = mu_v + eps[o] * expf(0.5f * lv_v);
    }
  }
}

// ---------------------------------------------------------------------------
// Decoder: per-type  z[16000,64] @ decW[t][64,256] + decb[t] -> recon
// grid = 3*500, block = 256 (2 row-strips x 4 col-groups per block)
// decW[t] pair-interleaved in LDS (64 KB); A (z rows) read from global (L2)
// ---------------------------------------------------------------------------
__global__ __launch_bounds__(256) void dec_kernel(
    const float* __restrict__ zb, const float* __restrict__ decW,
    const float* __restrict__ decb, float* __restrict__ out) {
  __shared__ v2f wl2[(kD / 2) * kDIN];  // 64 KB
  int blk = blockIdx.x;
  int t = blk / 500;
  int rb_ = blk % 500;
  int rowBase = rb_ * 32;  // within type
  stage_w_pairs(wl2, decW + (size_t)t * kD * kDIN, kD / 2, kDIN);
  __syncthreads();

  int wave = threadIdx.x >> 5, lane = threadIdx.x & 31;
  int rs = wave >> 2, cg = wave & 3;  // row strip, col group (64 cols each)
  int hi = lane >> 4, ln = lane & 15;
  const float* az = zb + ((size_t)t * kNT + rowBase + rs * 16 + ln) * kD;

  v8f c0 = {0,0,0,0,0,0,0,0}, c1 = c0, c2 = c0, c3 = c0;
  for (int k = 0; k < kD; k += 4) {
    v2f a = *(const v2f*)(az + k + 2 * hi);
    const v2f* wp = wl2 + (k / 2 + hi) * kDIN + cg * 64;
    v2f b0 = wp[ln];
    v2f b1 = wp[16 + ln];
    v2f b2 = wp[32 + ln];
    v2f b3 = wp[48 + ln];
    c0 = wmma4(a, b0, c0); c1 = wmma4(a, b1, c1);
    c2 = wmma4(a, b2, c2); c3 = wmma4(a, b3, c3);
  }
  const float* B = decb + t * kDIN + cg * 64;
  float* O = out + (size_t)t * kNT * kDIN;
  int mrow = rowBase + rs * 16 + 8 * hi;
#pragma unroll
  for (int r = 0; r < 8; ++r) {
    size_t o = (size_t)(mrow + r) * kDIN + cg * 64;
    O[o + ln]      = c0[r] + B[ln];
    O[o + 16 + ln] = c1[r] + B[16 + ln];
    O[o + 32 + ln] = c2[r] + B[32 + ln];
    O[o + 48 + ln] = c3[r] + B[48 + ln];
  }
}

// ---------------------------------------------------------------------------
// Host orchestration
// ---------------------------------------------------------------------------
extern "C" void kernel_launch(void* const* d_in, const int* in_sizes, int n_in,
                              void* d_out, int out_size, void* d_ws,
                              size_t ws_size, hipStream_t stream) {
  const float* x_rna   = (const float*)d_in[0];
  const float* x_meth  = (const float*)d_in[1];
  const float* x_snp   = (const float*)d_in[2];
  const float* eps     = (const float*)d_in[3];
  const float* edist   = (const float*)d_in[4];
  const float* encW    = (const float*)d_in[5];
  const float* encb    = (const float*)d_in[6];
  const float* decW    = (const float*)d_in[7];
  const float* decb    = (const float*)d_in[8];
  const float* qW      = (const float*)d_in[9];
  const float* qb      = (const float*)d_in[10];
  const float* kW      = (const float*)d_in[11];
  const float* kb      = (const float*)d_in[12];
  const float* vW      = (const float*)d_in[13];
  const float* vb      = (const float*)d_in[14];
  const float* rel_q   = (const float*)d_in[15];
  const float* rel_k   = (const float*)d_in[16];
  const float* rel_v   = (const float*)d_in[17];
  const float* skn     = (const float*)d_in[18];
  const float* svn     = (const float*)d_in[19];
  const float* rel_b   = (const float*)d_in[20];
  const float* d_alpha = (const float*)d_in[21];
  const float* d_tau   = (const float*)d_in[22];
  const float* skip    = (const float*)d_in[23];
  const float* ln_g    = (const float*)d_in[24];
  const float* ln_b    = (const float*)d_in[25];
  const float* muW     = (const float*)d_in[26];
  const float* mub     = (const float*)d_in[27];
  const float* lvW     = (const float*)d_in[28];
  const float* lvb     = (const float*)d_in[29];
  // d_in[30] node_type (implied by block ordering), d_in[31..33] edges
  const int* eidx  = (const int*)d_in[31];
  const int* etype = (const int*)d_in[32];
  const int* esign = (const int*)d_in[33];

  float* ws = (float*)d_ws;
  float* h      = ws + WS_H;
  float* Qb     = ws + WS_Q;
  float* Kb     = ws + WS_K;
  float* Vb     = ws + WS_V;
  float* score  = ws + WS_SCORE;
  unsigned* mk  = (unsigned*)(ws + WS_MKEY);
  float* Zb     = ws + WS_Z;
  float* outun  = ws + WS_OUTUN;
  float* zb     = ws + WS_ZB;
  float* out    = (float*)d_out;

  enc_kernel<<<kT * 125, 256, 0, stream>>>(x_rna, x_meth, x_snp, encW, encb, h);

  for (int l = 0; l < kL; ++l) {
    qkv_kernel<<<kT * 125, 256, 0, stream>>>(h, qW, qb, kW, kb, vW, vb, l, Qb,
                                             Kb, Vb);
    // zero mkey + Z + out_un (contiguous: N*H + N*H + N*D = 3,456,000 floats)
    fill0_kernel<<<13500, 256, 0, stream>>>(ws + WS_MKEY, 3456000);
    edgeA_kernel<<<(kE * kH) / 256, 256, 0, stream>>>(
        eidx, etype, esign, edist, Qb, Kb, rel_q + (size_t)l * kR * kH * kDK,
        rel_k + (size_t)l * kR * kH * kDK, skn + (size_t)l * kH * kDK,
        rel_b + (size_t)l * kR * kH, d_alpha + l, d_tau + l, score, mk);
    edgeB_kernel<<<(kE * kH) / 256, 256, 0, stream>>>(
        eidx, etype, esign, Vb, rel_v + (size_t)l * kR * kH * kDK,
        svn + (size_t)l * kH * kDK, score, mk, Zb, outun);
    nodeFin_kernel<<<kN / 8, 256, 0, stream>>>(outun, Zb, h, skip + l * kT,
                                               ln_g + (size_t)l * kT * kD,
                                               ln_b + (size_t)l * kT * kD);
  }

  mulvz_kernel<<<kN / 128, 256, 0, stream>>>(h, muW, mub, lvW, lvb, eps, out,
                                             zb);
  dec_kernel<<<kT * 500, 256, 0, stream>>>(zb, decW, decb, out);
}